// VPOS_Injection_60009283060004
// MI455X (gfx1250) — compile-verified
//
#include <hip/hip_runtime.h>
#include <math.h>

#define B_   4
#define C_   512
#define CK_  256
#define H_   64
#define W_   64
#define N_   4096
#define WP_  16

typedef float v8f __attribute__((ext_vector_type(8)));
typedef float v2f __attribute__((ext_vector_type(2)));
typedef int   v4i __attribute__((vector_size(16)));

#define AS1 __attribute__((address_space(1)))
#define AS3 __attribute__((address_space(3)))

#if defined(__has_builtin)
# if __has_builtin(__builtin_amdgcn_global_load_async_to_lds_b128) && \
     __has_builtin(__builtin_amdgcn_s_wait_asynccnt)
#  define USE_ASYNC_LDS 1
# endif
#endif
#ifndef USE_ASYNC_LDS
# define USE_ASYNC_LDS 0
#endif

// ---------------------------------------------------------------------------
// Kernel A: per (b,c): xsum[w] = sum_h x[b,c,h,w]; then x1d (group-of-4 mean
// + positional encoding) and 16->64 linear interpolation -> row[b,c,w].
// ---------------------------------------------------------------------------
__global__ __launch_bounds__(256) void kA_reduce(
    const float* __restrict__ x, const float* __restrict__ pe_table,
    const int* __restrict__ pos, float* __restrict__ xsum,
    float* __restrict__ rowv)
{
  __shared__ float part[4][64];
  __shared__ float xs[64];
  __shared__ float x1d[16];
  const int bc  = blockIdx.x;       // b*512 + c
  const int b   = bc >> 9;
  const int c   = bc & 511;
  const int tid = threadIdx.x;
  const int w   = tid & 63;
  const int hg  = tid >> 6;         // 0..3
  const float* xp = x + (size_t)bc * H_ * W_;
  float s = 0.f;
#pragma unroll
  for (int j = 0; j < 16; ++j) s += xp[(hg + 4 * j) * W_ + w];
  part[hg][w] = s;
  __syncthreads();
  if (tid < 64) {
    float v = part[0][w] + part[1][w] + part[2][w] + part[3][w];
    xsum[(size_t)bc * W_ + w] = v;
    xs[w] = v;
  }
  __syncthreads();
  if (tid < 16) {
    const int j = tid;
    float v = (xs[4*j] + xs[4*j+1] + xs[4*j+2] + xs[4*j+3]) * (1.0f / 256.0f);
    const int p = pos[b * (H_ * W_) + 4 * j] >> 3;  // pos[b,0,4j] // 8
    v += pe_table[p * C_ + c];
    x1d[j] = v;
  }
  __syncthreads();
  if (tid < 64) {
    const float src = (float)w * 15.0f / 63.0f;
    int i0 = (int)floorf(src);
    int i1 = i0 + 1; if (i1 > 15) i1 = 15;
    const float t = src - (float)i0;
    rowv[(size_t)bc * W_ + w] = x1d[i0] * (1.0f - t) + x1d[i1] * t;
  }
}

// ---------------------------------------------------------------------------
// Kernel B1: khat[b,o,w] = Wk[o,:] . row[b,:,w] + bk[o]
// ---------------------------------------------------------------------------
__global__ __launch_bounds__(256) void kB_khat(
    const float* __restrict__ Wk, const float* __restrict__ bk,
    const float* __restrict__ rowv, float* __restrict__ khat)
{
  const int t = blockIdx.x * 256 + threadIdx.x;   // ((b*256+o)*64+w)
  const int b = t >> 14;
  const int rem = t & 16383;
  const int o = rem >> 6;
  const int w = rem & 63;
  const float* wk = Wk + o * C_;
  const float* rp = rowv + (size_t)b * C_ * W_ + w;
  float s = bk[o];
  for (int cc = 0; cc < C_; ++cc) s += wk[cc] * rp[cc * W_];
  khat[t] = s;
}

// ---------------------------------------------------------------------------
// Kernel B2: pair-packed vmean, w-major:
//   vmTp[b][k2][c] = float2{ vmean[c][2k2], vmean[c][2k2+1] }
//   vmean[c][w] = (Wv[c,:] . xsum[b,:,w]) / H + bv[c]
// ---------------------------------------------------------------------------
__global__ __launch_bounds__(256) void kB_vmean(
    const float* __restrict__ Wv, const float* __restrict__ bv,
    const float* __restrict__ xsum, float* __restrict__ vmTp)
{
  const int t = blockIdx.x * 256 + threadIdx.x;   // ((b*32+k2)*512+c)
  const int b  = t >> 14;
  const int rem = t & 16383;
  const int k2 = rem >> 9;
  const int c  = rem & 511;
  const int w0 = 2 * k2;
  const float* wv = Wv + c * C_;
  const float* xp = xsum + (size_t)b * C_ * W_ + w0;
  float s0 = 0.f, s1 = 0.f;
  for (int cc = 0; cc < C_; ++cc) {
    const float wvc = wv[cc];
    s0 += wvc * xp[cc * W_];
    s1 += wvc * xp[cc * W_ + 1];
  }
  const float bb = bv[c];
  vmTp[2 * (size_t)t]     = s0 * (1.0f / 64.0f) + bb;
  vmTp[2 * (size_t)t + 1] = s1 * (1.0f / 64.0f) + bb;
}

// ---------------------------------------------------------------------------
// Kernel B3: pair-packed M:  Mp[b][c2][w] = float2{ M[2c2][w], M[2c2+1][w] }
//   M[c][w] = sum_o Wq[o,c] * khat[b,o,w]   (energy = x^T M + c0)
// ---------------------------------------------------------------------------
__global__ __launch_bounds__(256) void kB_M(
    const float* __restrict__ Wq, const float* __restrict__ khat,
    float* __restrict__ Mp)
{
  const int t = blockIdx.x * 256 + threadIdx.x;   // ((b*256+c2)*64+w)
  const int b  = t >> 14;
  const int rem = t & 16383;
  const int c2 = rem >> 6;
  const int w  = rem & 63;
  const float* kp = khat + (size_t)b * CK_ * W_ + w;
  float s0 = 0.f, s1 = 0.f;
  for (int o = 0; o < CK_; ++o) {
    const float kv = kp[o * W_];
    s0 += Wq[o * C_ + 2 * c2]     * kv;
    s1 += Wq[o * C_ + 2 * c2 + 1] * kv;
  }
  Mp[2 * (size_t)t]     = s0;
  Mp[2 * (size_t)t + 1] = s1;
}

// ---------------------------------------------------------------------------
// Kernel B4: c0[b,w] = bq . khat[b,:,w]
// ---------------------------------------------------------------------------
__global__ __launch_bounds__(256) void kB_c0(
    const float* __restrict__ bq, const float* __restrict__ khat,
    float* __restrict__ c0w)
{
  const int t = threadIdx.x;   // b*64 + w
  const int b = t >> 6;
  const int w = t & 63;
  const float* kp = khat + (size_t)b * CK_ * W_ + w;
  float s = 0.f;
  for (int o = 0; o < CK_; ++o) s += bq[o] * kp[o * W_];
  c0w[t] = s;
}

// ---------------------------------------------------------------------------
// Kernel C: fused main kernel. 8 waves/block, 16 query rows per wave.
//  Stage 0: async-copy pair-packed M (128 KB) into LDS (ASYNCcnt path)
//  Phase 1: e(16x64) = x_tile^T(16x512) @ M(512x64)      [WMMA f32 16x16x4,
//           A: global b32x2, B: ds_load_b64 from LDS]
//  Softmax over 64 w per row (shfl_xor width-16, wave32); attn kept in regs
//  LDS reuse: M region is dead -> attn (pad-66 rows) written into same buffer
//  Phase 2: out(512x16) = vmean(512x64) @ attn^T(64x16)  [WMMA f32 16x16x4,
//           A: global b64 (pair-packed), B: ds_load_b64]
//  Store: out = gamma*out + x (residual fused)
// ---------------------------------------------------------------------------
__global__ __launch_bounds__(256) void kC_attn(
    const float* __restrict__ x, const float* __restrict__ Mp,
    const float* __restrict__ c0w, const float* __restrict__ vmTp,
    const float* __restrict__ gammap, float* __restrict__ outp)
{
  // 32768 floats = 128 KB.  Phase 1: pair-packed M.  Phase 2: attn (8*16*66).
  __shared__ float smem[C_ * W_];
  const int b    = blockIdx.y;
  const int tid  = threadIdx.x;
  const int wave = tid >> 5;
  const int lane = tid & 31;
  const int half = lane >> 4;     // 0 = K 0/1, 1 = K 2/3 in WMMA A/B
  const int l16  = lane & 15;
  const int n0   = blockIdx.x * 128 + wave * 16;
  const float* xb  = x    + (size_t)b * C_ * N_;
  const float* Mpb = Mp   + (size_t)b * C_ * W_;
  const float  g   = gammap[0];

  // -------- Stage 0: cooperative copy of Mp into LDS (async if available) ---
#if USE_ASYNC_LDS
#pragma unroll 4
  for (int i = tid; i < (C_ * W_) / 4; i += 256) {
    __builtin_amdgcn_global_load_async_to_lds_b128(
        (AS1 v4i*)(Mpb + 4 * (size_t)i), (AS3 v4i*)(smem + 4 * i), 0, 0);
  }
  __builtin_amdgcn_s_wait_asynccnt(0);
#else
#pragma unroll 4
  for (int i = tid; i < (C_ * W_) / 4; i += 256) {
    const float4 v = ((const float4*)Mpb)[i];
    ((float4*)smem)[i] = v;
  }
#endif
  __syncthreads();

  // -------- Phase 1: energy e[16 rows n][64 cols w] --------
  v8f acc[4] = {};
  for (int k = 0; k < C_; k += 4) {
    const int kk = k + 2 * half;
    v2f a;
    a.x = xb[(size_t)kk       * N_ + n0 + l16];
    a.y = xb[(size_t)(kk + 1) * N_ + n0 + l16];
    if ((k & 31) == 0 && (k + 32) < C_)
      __builtin_prefetch(&xb[(size_t)(k + 32) * N_ + n0 + l16], 0, 0);
#pragma unroll
    for (int wt = 0; wt < 4; ++wt) {
      const int wc = wt * 16 + l16;
      const v2f bm = *(const v2f*)(smem + 2 * ((kk >> 1) * W_ + wc));
      acc[wt] = __builtin_amdgcn_wmma_f32_16x16x4_f32(
          false, a, false, bm, (short)0, acc[wt], false, false);
    }
  }

  // -------- Softmax over 64 w per row; keep attn in registers --------
  float c0v[4];
#pragma unroll
  for (int wt = 0; wt < 4; ++wt) c0v[wt] = c0w[b * W_ + wt * 16 + l16];
  float att[8][4];
#pragma unroll
  for (int j = 0; j < 8; ++j) {
    float e0 = acc[0][j] + c0v[0];
    float e1 = acc[1][j] + c0v[1];
    float e2 = acc[2][j] + c0v[2];
    float e3 = acc[3][j] + c0v[3];
    float m = fmaxf(fmaxf(e0, e1), fmaxf(e2, e3));
#pragma unroll
    for (int off = 1; off < 16; off <<= 1)
      m = fmaxf(m, __shfl_xor(m, off, 16));
    float p0 = __expf(e0 - m), p1 = __expf(e1 - m);
    float p2 = __expf(e2 - m), p3 = __expf(e3 - m);
    float s = p0 + p1 + p2 + p3;
#pragma unroll
    for (int off = 1; off < 16; off <<= 1)
      s += __shfl_xor(s, off, 16);
    const float inv = 1.0f / s;
    att[j][0] = p0 * inv; att[j][1] = p1 * inv;
    att[j][2] = p2 * inv; att[j][3] = p3 * inv;
  }

  // All waves done reading M from LDS -> reuse buffer for attn.
  __syncthreads();
  float* arow = smem + wave * 16 * 66;   // pad-66 rows: conflict-free b64
  {
    const int r = 8 * half;              // D-layout row base
#pragma unroll
    for (int j = 0; j < 8; ++j) {
#pragma unroll
      for (int wt = 0; wt < 4; ++wt)
        arow[(r + j) * 66 + wt * 16 + l16] = att[j][wt];
    }
  }
  __syncthreads();

  // -------- Phase 2: out[c, n] = sum_w vmean[c,w] * attn[n,w] --------
  const float* vb = vmTp + (size_t)b * W_ * C_;   // pair-packed [k2][c]
  float*       ob = outp + (size_t)b * C_ * N_;
  for (int ct = 0; ct < 32; ++ct) {
    const int c0r = ct * 16;
    v8f oa = {};
#pragma unroll
    for (int kw = 0; kw < W_; kw += 4) {
      const int kk = kw + 2 * half;
      const v2f a  = *(const v2f*)(vb + 2 * ((size_t)(kk >> 1) * C_ + c0r + l16));
      const v2f bt = *(const v2f*)(arow + l16 * 66 + kk);
      oa = __builtin_amdgcn_wmma_f32_16x16x4_f32(
          false, a, false, bt, (short)0, oa, false, false);
    }
#pragma unroll
    for (int j = 0; j < 8; ++j) {
      const int c = c0r + j + 8 * half;
      const size_t idx = (size_t)c * N_ + n0 + l16;
      ob[idx] = g * oa[j] + xb[idx];
    }
  }
}

// ---------------------------------------------------------------------------
extern "C" void kernel_launch(void* const* d_in, const int* in_sizes, int n_in,
                              void* d_out, int out_size, void* d_ws,
                              size_t ws_size, hipStream_t stream) {
  const float* x     = (const float*)d_in[0];
  const float* Wq    = (const float*)d_in[1];
  const float* bq    = (const float*)d_in[2];
  const float* Wk    = (const float*)d_in[3];
  const float* bk    = (const float*)d_in[4];
  const float* Wv    = (const float*)d_in[5];
  const float* bv    = (const float*)d_in[6];
  const float* gamma = (const float*)d_in[7];
  const float* pe    = (const float*)d_in[8];
  const int*   pos   = (const int*)d_in[9];
  float* out = (float*)d_out;

  float* ws   = (float*)d_ws;
  float* xsum = ws;                     // B*C*W   = 131072
  float* rowv = xsum + B_ * C_ * W_;    // 131072
  float* khat = rowv + B_ * C_ * W_;    // B*CK*W  = 65536
  float* vmTp = khat + B_ * CK_ * W_;   // B*W*C   = 131072 (pair-packed)
  float* Mp   = vmTp + B_ * W_ * C_;    // B*C*W   = 131072 (pair-packed)
  float* c0w  = Mp   + B_ * C_ * W_;    // B*W     = 256
  // total ~2.36 MB of workspace

  kA_reduce<<<dim3(B_ * C_), dim3(256), 0, stream>>>(x, pe, pos, xsum, rowv);
  kB_khat <<<dim3(B_ * CK_ * W_ / 256), dim3(256), 0, stream>>>(Wk, bk, rowv, khat);
  kB_vmean<<<dim3(B_ * W_ * C_ / 512), dim3(256), 0, stream>>>(Wv, bv, xsum, vmTp);
  kB_M    <<<dim3(B_ * CK_ * W_ / 256), dim3(256), 0, stream>>>(Wq, khat, Mp);
  kB_c0   <<<dim3(1), dim3(256), 0, stream>>>(bq, khat, c0w);
  kC_attn <<<dim3(N_ / 128, B_), dim3(256), 0, stream>>>(x, Mp, c0w, vmTp, gamma, out);
}